// MoECombiner_39685497815990
// MI455X (gfx1250) — compile-verified
//
#include <hip/hip_runtime.h>

typedef __attribute__((ext_vector_type(2))) float v2f;
typedef __attribute__((ext_vector_type(8))) float v8f;

#define NUM_EXPERTS 64
#define NUM_IMAGES  16384
#define D_MODEL     4096

// One wave computes a 16(M) x 64(N) tile of out = gates[16384x64] * expert[64x4096].
// fp32 WMMA 16x16x4, K=64 -> 16 chained k-steps, A reused across 4 N-subtiles.
__global__ __launch_bounds__(256) void moe_combine_wmma_f32(
    const float* __restrict__ expert,  // [64, 4096]   d_in[0]
    const float* __restrict__ gates,   // [16384, 64]  d_in[1]
    float* __restrict__ out)           // [16384, 4096]
{
    const int lane  = threadIdx.x & 31;
    const int wave  = threadIdx.x >> 5;
    const int w     = blockIdx.x * 8 + wave;   // global wave id, 0..65535
    const int tileM = w >> 6;                  // 0..1023 -> rows tileM*16 ..
    const int tileN = (w & 63) << 6;           // 0,64,..,4032 -> col base

    // ---- A-matrix (gates) addressing: 16x4 f32 per k-step ----
    // lane L: row = L%16 ; VGPR pair holds K = k0 + 2*(L/16) + {0,1}
    const int rowA = lane & 15;
    const int ksel = (lane >> 4) << 1;         // 0 or 2
    const float* gRow = gates + (tileM * 16 + rowA) * NUM_EXPERTS + ksel;

    // ---- B-matrix (expert) addressing: 4x16 f32 per k-step ----
    // VGPR j, lanes 0-15: K=k0+j ; lanes 16-31: K=k0+2+j ; N = n0 + lane%16
    const int ncol = lane & 15;
    const float* eBase = expert + tileN + ncol;

    v8f acc0 = {}, acc1 = {}, acc2 = {}, acc3 = {};

#pragma unroll
    for (int ks = 0; ks < 16; ++ks) {
        const int k0 = ks * 4;
        const v2f a = *(const v2f*)(gRow + k0);

        const float* ek = eBase + (size_t)(k0 + ksel) * D_MODEL;
        v2f b0, b1, b2, b3;
        b0.x = ek[0];   b0.y = ek[D_MODEL + 0];
        b1.x = ek[16];  b1.y = ek[D_MODEL + 16];
        b2.x = ek[32];  b2.y = ek[D_MODEL + 32];
        b3.x = ek[48];  b3.y = ek[D_MODEL + 48];

        // 8 args: (neg_a, A, neg_b, B, c_mod, C, reuse_a, reuse_b)
        acc0 = __builtin_amdgcn_wmma_f32_16x16x4_f32(false, a, false, b0, (short)0, acc0, false, false);
        acc1 = __builtin_amdgcn_wmma_f32_16x16x4_f32(false, a, false, b1, (short)0, acc1, false, false);
        acc2 = __builtin_amdgcn_wmma_f32_16x16x4_f32(false, a, false, b2, (short)0, acc2, false, false);
        acc3 = __builtin_amdgcn_wmma_f32_16x16x4_f32(false, a, false, b3, (short)0, acc3, false, false);
    }

    // ---- C/D layout: VGPR r -> row r (lanes 0-15) / row 8+r (lanes 16-31), col = lane%16 ----
    const int rowBase = tileM * 16 + ((lane >> 4) << 3);
    float* o = out + (size_t)rowBase * D_MODEL + tileN + ncol;
#pragma unroll
    for (int r = 0; r < 8; ++r) {
        float* orow = o + (size_t)r * D_MODEL;
        // Non-temporal: the 268MB output stream must not evict L2-resident inputs.
        __builtin_nontemporal_store(acc0[r], orow + 0);
        __builtin_nontemporal_store(acc1[r], orow + 16);
        __builtin_nontemporal_store(acc2[r], orow + 32);
        __builtin_nontemporal_store(acc3[r], orow + 48);
    }
}

extern "C" void kernel_launch(void* const* d_in, const int* in_sizes, int n_in,
                              void* d_out, int out_size, void* d_ws, size_t ws_size,
                              hipStream_t stream) {
    const float* expert = (const float*)d_in[0];  // [64, 4096]
    const float* gates  = (const float*)d_in[1];  // [16384, 64]
    float* out          = (float*)d_out;          // [16384, 4096]

    // 1024 M-tiles * 64 N64-tiles = 65536 waves; 8 waves (256 threads) per block.
    const int numBlocks = (NUM_IMAGES / 16) * (D_MODEL / 64) / 8;  // 8192
    moe_combine_wmma_f32<<<numBlocks, 256, 0, stream>>>(expert, gates, out);
}